// FLoSP_37151467111178
// MI455X (gfx1250) — compile-verified
//
#include <hip/hip_runtime.h>

// ---------------- problem constants ----------------
#define B_    2
#define C_    200
#define D3    16
#define H3    60
#define W3    80
#define P3    (D3*H3*W3)        // 76800 plane elems of 3D vol
#define H2c   120
#define W2c   160
#define P2    (H2c*W2c)         // 19200 plane elems of 2D map
#define SX    60
#define SY    36
#define SZ    60
#define NVOX  (SX*SY*SZ)        // 129600
#define NOUT  (B_*C_*NVOX)      // 51,840,000
#define ZTILE 30
#define NTAPS 12

// Per-voxel tap plane-offsets (no channel term) + weights.
// taps 0..7  -> 3D volume  (offset into (D,H,W) plane-index space)
// taps 8..11 -> 2D map     (offset into (H2,W2) plane-index space)
__device__ __forceinline__ void voxel_taps(float px, float py, float pz,
                                           int* __restrict__ off,
                                           float* __restrict__ w) {
  // ---- 3D trilinear ----
  float gx = (px + 1.0f) * 0.5f * (float)(W3 - 1);
  float gy = (py + 1.0f) * 0.5f * (float)(H3 - 1);
  float gz = (pz + 1.0f) * 0.5f * (float)(D3 - 1);
  float fx = floorf(gx), fy = floorf(gy), fz = floorf(gz);
  float wx = gx - fx, wy = gy - fy, wz = gz - fz;
  int x0 = (int)fx, y0 = (int)fy, z0 = (int)fz;
#pragma unroll
  for (int dz = 0; dz < 2; ++dz)
#pragma unroll
    for (int dy = 0; dy < 2; ++dy)
#pragma unroll
      for (int dx = 0; dx < 2; ++dx) {
        int t = (dz << 2) | (dy << 1) | dx;
        int xi = x0 + dx, yi = y0 + dy, zi = z0 + dz;
        bool valid = (xi >= 0) & (xi < W3) & (yi >= 0) & (yi < H3) &
                     (zi >= 0) & (zi < D3);
        int xc = min(max(xi, 0), W3 - 1);
        int yc = min(max(yi, 0), H3 - 1);
        int zc = min(max(zi, 0), D3 - 1);
        off[t] = (zc * H3 + yc) * W3 + xc;
        float ww = (dx ? wx : 1.0f - wx) * (dy ? wy : 1.0f - wy) *
                   (dz ? wz : 1.0f - wz);
        w[t] = valid ? ww : 0.0f;
      }
  // ---- 2D bilinear (uses px,py only) ----
  float hx = (px + 1.0f) * 0.5f * (float)(W2c - 1);
  float hy = (py + 1.0f) * 0.5f * (float)(H2c - 1);
  float fhx = floorf(hx), fhy = floorf(hy);
  float ux = hx - fhx, uy = hy - fhy;
  int u0 = (int)fhx, v0 = (int)fhy;
#pragma unroll
  for (int dy = 0; dy < 2; ++dy)
#pragma unroll
    for (int dx = 0; dx < 2; ++dx) {
      int t = 8 + (dy << 1) + dx;
      int xi = u0 + dx, yi = v0 + dy;
      bool valid = (xi >= 0) & (xi < W2c) & (yi >= 0) & (yi < H2c);
      int xc = min(max(xi, 0), W2c - 1);
      int yc = min(max(yi, 0), H2c - 1);
      off[t] = yc * W2c + xc;
      float ww = (dx ? ux : 1.0f - ux) * (dy ? uy : 1.0f - uy);
      w[t] = valid ? ww : 0.0f;
    }
}

// -------- phase 1: fused add + transpose (b,C,P) -> (b,P,C) --------
__global__ __launch_bounds__(256) void addtrans_kernel(
    const float* __restrict__ a, const float* __restrict__ bsrc,
    float* __restrict__ out, int P) {
  __shared__ float tile[32][33];
  int p0 = blockIdx.x * 32;
  int c0 = blockIdx.y * 32;
  int b = blockIdx.z;
  const float* ab = a + (size_t)b * C_ * P;
  const float* bb = bsrc + (size_t)b * C_ * P;
  float* ob = out + (size_t)b * P * C_;
  int tx = threadIdx.x;  // 0..31
  int ty = threadIdx.y;  // 0..7
#pragma unroll
  for (int r = 0; r < 4; ++r) {
    int c = c0 + ty + r * 8;
    int p = p0 + tx;
    if (c < C_ && p < P)
      tile[ty + r * 8][tx] = ab[(size_t)c * P + p] + bb[(size_t)c * P + p];
  }
  __syncthreads();
#pragma unroll
  for (int r = 0; r < 4; ++r) {
    int p = p0 + ty + r * 8;
    int c = c0 + tx;
    if (c < C_ && p < P) ob[(size_t)p * C_ + c] = tile[tx][ty + r * 8];
  }
}

// -------- phase 2: gather from channel-last staging, async-to-LDS pipelined --
// Each z needs 12 taps x 200 channels = 12 contiguous 800B runs.
// Issued as B128 asyncs: 12 taps x 50 j-slots of 16B, padded to 64 j-slots so
// every 64-thread group (2 waves) owns taps t with (t&3)==g  ->  EXACTLY 3
// async instructions per wave per z (uniform, so s_wait_asynccnt immediates
// are valid for every wave).
__global__ __launch_bounds__(256) void flosp_gather_kernel(
    const float* __restrict__ s3, const float* __restrict__ s2,
    const float* __restrict__ pp, float* __restrict__ out) {
  __shared__ __align__(16) float ltap[2 * NTAPS * C_];  // 19200 B
  __shared__ float lout[ZTILE * (C_ + 1)];              // 24120 B

  const int x = blockIdx.x;   // 0..59
  const int y = blockIdx.y;   // 0..35
  const int b = blockIdx.z;   // 0..1
  const int tid = (int)threadIdx.x;
  const int c = tid;          // channel lane for compute, active if c < C_
  const int g = tid >> 6;     // 64-thread group 0..3 (uniform per 2 waves)
  const int j = tid & 63;     // 16B slot within tap

  const float* s3b = s3 + (size_t)b * P3 * C_;
  const float* s2b = s2 + (size_t)b * P2 * C_;
  float* outb = out + (size_t)b * (size_t)C_ * NVOX + (size_t)x * (SY * SZ) +
                (size_t)y * SZ;

  // issue one z worth of gathers (12 x 800B) into buffer `buf`
  auto issue = [&](int z, int buf) {
    int vox = x * (SZ * SY) + z * SY + y;
    const float* gpp = pp + ((size_t)b * NVOX + vox) * 3;
    int off[NTAPS];
    float w[NTAPS];
    voxel_taps(gpp[0], gpp[1], gpp[2], off, w);
#pragma unroll
    for (int t = 0; t < NTAPS; ++t) {
      if ((t & 3) == g && j < 50) {  // t compile-time, branch wave-uniform
        const float* src =
            ((t < 8) ? s3b : s2b) + (size_t)off[t] * C_ + j * 4;
        unsigned lofs =
            (unsigned)(size_t)(&ltap[buf * (NTAPS * C_) + t * C_ + j * 4]);
        unsigned long long ga = (unsigned long long)(size_t)src;
        asm volatile("global_load_async_to_lds_b128 %0, %1, off"
                     :
                     : "v"(lofs), "v"(ga)
                     : "memory");
      }
    }
  };

  for (int zt = 0; zt < SZ; zt += ZTILE) {
    issue(zt, 0);  // prime the pipeline
    for (int zi = 0; zi < ZTILE; ++zi) {
      const int z = zt + zi;
      const int buf = zi & 1;
      const bool havenext = (zi + 1 < ZTILE);
      if (havenext) {
        issue(z + 1, buf ^ 1);
        // each wave: 3 outstanding for z + 3 for z+1; <=3 means z complete
        asm volatile("s_wait_asynccnt 3" ::: "memory");
      } else {
        asm volatile("s_wait_asynccnt 0" ::: "memory");  // drain
      }
      __syncthreads();
      if (c < C_) {
        int vox = x * (SZ * SY) + z * SY + y;
        const float* gpp = pp + ((size_t)b * NVOX + vox) * 3;
        int off[NTAPS];
        float w[NTAPS];
        voxel_taps(gpp[0], gpp[1], gpp[2], off, w);
        float acc = 0.0f;
#pragma unroll
        for (int t = 0; t < NTAPS; ++t)
          acc += w[t] * ltap[buf * (NTAPS * C_) + t * C_ + c];
        lout[zi * (C_ + 1) + c] = acc;
      }
      __syncthreads();  // protect ltap[buf] reuse + publish lout row
    }
    // transposed, coalesced flush: lanes run along z (output innermost dim)
    for (int i = tid; i < ZTILE * C_; i += 256) {
      int cc = i / ZTILE;
      int zl = i % ZTILE;
      outb[(size_t)cc * NVOX + (zt + zl)] = lout[zl * (C_ + 1) + cc];
    }
    __syncthreads();  // before next tile reuses LDS
  }
}

// -------- fallback: direct gather from original NCHW layout (ws too small) --
__global__ __launch_bounds__(256) void flosp_direct_kernel(
    const float* __restrict__ x3a, const float* __restrict__ x3b,
    const float* __restrict__ x2a, const float* __restrict__ x2b,
    const float* __restrict__ pp, float* __restrict__ out) {
  int gid = blockIdx.x * blockDim.x + threadIdx.x;
  if (gid >= NOUT) return;
  int z = gid % SZ;
  int t1 = gid / SZ;
  int yy = t1 % SY;
  int t2 = t1 / SY;
  int xx = t2 % SX;
  int t3 = t2 / SX;
  int cc = t3 % C_;
  int bb = t3 / C_;
  int vox = xx * (SZ * SY) + z * SY + yy;
  const float* g = pp + ((size_t)bb * NVOX + vox) * 3;
  int off[NTAPS];
  float w[NTAPS];
  voxel_taps(g[0], g[1], g[2], off, w);
  const float* a3 = x3a + ((size_t)bb * C_ + cc) * P3;
  const float* b3 = x3b + ((size_t)bb * C_ + cc) * P3;
  const float* a2 = x2a + ((size_t)bb * C_ + cc) * P2;
  const float* b2 = x2b + ((size_t)bb * C_ + cc) * P2;
  float acc = 0.0f;
#pragma unroll
  for (int t = 0; t < 8; ++t) acc += w[t] * (a3[off[t]] + b3[off[t]]);
#pragma unroll
  for (int t = 8; t < NTAPS; ++t) acc += w[t] * (a2[off[t]] + b2[off[t]]);
  out[gid] = acc;
}

extern "C" void kernel_launch(void* const* d_in, const int* in_sizes, int n_in,
                              void* d_out, int out_size, void* d_ws,
                              size_t ws_size, hipStream_t stream) {
  (void)in_sizes; (void)n_in; (void)out_size;
  const float* x3a = (const float*)d_in[0];
  const float* x3b = (const float*)d_in[1];
  const float* x2a = (const float*)d_in[2];
  const float* x2b = (const float*)d_in[3];
  const float* pp  = (const float*)d_in[4];
  float* out = (float*)d_out;

  const size_t s3_elems = (size_t)B_ * P3 * C_;  // 30,720,000
  const size_t s2_elems = (size_t)B_ * P2 * C_;  //  7,680,000
  const size_t need = (s3_elems + s2_elems) * sizeof(float);  // 153.6 MB

  if (d_ws != nullptr && ws_size >= need) {
    float* s3 = (float*)d_ws;
    float* s2 = s3 + s3_elems;
    // phase 1: fused add + NCHW->channel-last transpose
    addtrans_kernel<<<dim3((P3 + 31) / 32, (C_ + 31) / 32, B_), dim3(32, 8), 0,
                      stream>>>(x3a, x3b, s3, P3);
    addtrans_kernel<<<dim3((P2 + 31) / 32, (C_ + 31) / 32, B_), dim3(32, 8), 0,
                      stream>>>(x2a, x2b, s2, P2);
    // phase 2: pipelined async B128 gather + transposed coalesced store
    flosp_gather_kernel<<<dim3(SX, SY, B_), 256, 0, stream>>>(s3, s2, pp, out);
  } else {
    flosp_direct_kernel<<<(NOUT + 255) / 256, 256, 0, stream>>>(
        x3a, x3b, x2a, x2b, pp, out);
  }
}